// GCNModel_76510547411355
// MI455X (gfx1250) — compile-verified
//
#include <hip/hip_runtime.h>
#include <hip/hip_bf16.h>

typedef __attribute__((ext_vector_type(2))) float v2f;
typedef __attribute__((ext_vector_type(8))) float v8f;

static __device__ __forceinline__ void atomAddF(float* p, float v) {
    // relaxed agent-scope fadd -> global_atomic_add_f32 on gfx1250
    __hip_atomic_fetch_add(p, v, __ATOMIC_RELAXED, __HIP_MEMORY_SCOPE_AGENT);
}

static __device__ __forceinline__ float leaky(float v) {
    return (v > 0.0f) ? v : 0.01f * v;
}

// ---------------- degree / normalization ----------------
__global__ void fill1_kernel(float* __restrict__ p, int n) {
    int i = blockIdx.x * blockDim.x + threadIdx.x;
    if (i < n) p[i] = 1.0f;  // self loop contributes 1 to every node's degree
}

__global__ void deg_edges_kernel(const long long* __restrict__ dst, float* __restrict__ deg, int e) {
    int i = blockIdx.x * blockDim.x + threadIdx.x;
    if (i < e) atomAddF(&deg[(int)dst[i]], 1.0f);
}

__global__ void rsqrt_ip_kernel(float* __restrict__ p, int n) {
    int i = blockIdx.x * blockDim.x + threadIdx.x;
    if (i < n) p[i] = rsqrtf(p[i]);  // deg >= 1 always
}

// ------- fused dense GEMM via V_WMMA_F32_16X16X4_F32 -------
// Y[n x NC] = act(X)[n x K] @ W[K x NC], where act(x) = leaky(x + preBias[c])
// when preBias != null (fuses previous layer's bias+leaky into LDS staging).
// Epilogue additionally writes Agg = Y * dis[row]^2 (self-loop init of the
// scatter accumulator), so xw is never re-read for initialization.
// One block = 16 rows; wave w = columns [16w,16w+16); blockDim.x = 2*NC.
// Dynamic LDS: 16*(K+1) floats (pad 1/row -> conflict-free column reads).
__global__ void gemm16_wmma_fused_kernel(const float* __restrict__ X, const float* __restrict__ W,
                                         const float* __restrict__ preBias,  // nullable
                                         const float* __restrict__ dis,
                                         float* __restrict__ Y, float* __restrict__ Agg,
                                         int n, int K, int NC) {
    extern __shared__ float lds[];
    const int LK = K + 1;
    const int rowbase = blockIdx.x * 16;
    const int tid = threadIdx.x;

    // cooperatively stage 16 x K tile of act(X) (zero-pad OOB rows)
    for (int idx = tid; idx < 16 * K; idx += blockDim.x) {
        int r = idx / K;
        int c = idx - r * K;
        int gr = rowbase + r;
        float v = 0.0f;
        if (gr < n) {
            v = X[(size_t)gr * K + c];
            if (preBias) v = leaky(v + preBias[c]);
        }
        lds[r * LK + c] = v;
    }
    __syncthreads();

    const int lane = tid & 31;
    const int half = lane >> 4;   // 0: K pair {0,1}; 1: K pair {2,3}
    const int l16  = lane & 15;
    const int colbase = (tid >> 5) * 16;

    v8f acc = {};
    for (int k = 0; k < K; k += 4) {
        v2f a, b;
        // A 16x4: lanes 0-15 -> row l16, K = k+0,k+1 ; lanes 16-31 -> K = k+2,k+3
        const float* xr = &lds[l16 * LK + k + 2 * half];
        a.x = xr[0];
        a.y = xr[1];
        // B 4x16: v0 = K(k+2h), v1 = K(k+2h+1), column = colbase + l16
        const float* wr = &W[(size_t)(k + 2 * half) * NC + colbase + l16];
        b.x = wr[0];
        b.y = wr[NC];
        acc = __builtin_amdgcn_wmma_f32_16x16x4_f32(
            /*neg_a=*/false, a, /*neg_b=*/false, b,
            /*c_mod=*/(short)0, acc, /*reuse_a=*/false, /*reuse_b=*/false);
    }

    // C/D 16x16: VGPR i -> M = i (lanes 0-15) / i+8 (lanes 16-31), N = l16
#pragma unroll
    for (int i = 0; i < 8; ++i) {
        int r = rowbase + i + 8 * half;
        if (r < n) {
            float v = acc[i];
            size_t o = (size_t)r * NC + colbase + l16;
            Y[o] = v;
            float s = dis[r];        // uniform across half-wave -> broadcast
            Agg[o] = v * s * s;      // self-loop term initializes accumulator
        }
    }
}

// ---------------- edge scatter ----------------
// agg[dst][f..f+3] += xw[src][f..f+3] * dis[src]*dis[dst]
// one thread per (edge, 4 features): b128 gather + 4 hw fp32 atomics.
template <int F>
__global__ void scatter_kernel(const float* __restrict__ xw, const long long* __restrict__ src,
                               const long long* __restrict__ dst, const float* __restrict__ dis,
                               float* __restrict__ agg, int e) {
    constexpr int FQ = F / 4;
    int t = blockIdx.x * blockDim.x + threadIdx.x;
    if (t >= e * FQ) return;
    int ed = t / FQ;                // FQ power of two -> shift
    int f  = (t & (FQ - 1)) * 4;
    int s = (int)src[ed];
    int d = (int)dst[ed];
    float nm = dis[s] * dis[d];
    const float4 v = *(const float4*)(xw + (size_t)s * F + f);
    float* p = agg + (size_t)d * F + f;
    atomAddF(p + 0, v.x * nm);
    atomAddF(p + 1, v.y * nm);
    atomAddF(p + 2, v.z * nm);
    atomAddF(p + 3, v.w * nm);
}

// y = leaky_relu(agg + b, 0.01)  (final layer only; layer-1's is fused in gemm2)
template <int F>
__global__ void bias_relu_kernel(const float* __restrict__ agg, const float* __restrict__ b,
                                 float* __restrict__ y, int total) {
    int i = blockIdx.x * blockDim.x + threadIdx.x;
    if (i < total) y[i] = leaky(agg[i] + b[i & (F - 1)]);
}

static inline int cdiv(int a, int b) { return (a + b - 1) / b; }

extern "C" void kernel_launch(void* const* d_in, const int* in_sizes, int n_in,
                              void* d_out, int out_size, void* d_ws, size_t ws_size,
                              hipStream_t stream) {
    const float*     X  = (const float*)d_in[0];
    const long long* EI = (const long long*)d_in[1];
    const float*     W1 = (const float*)d_in[2];
    const float*     b1 = (const float*)d_in[3];
    const float*     W2 = (const float*)d_in[4];
    const float*     b2 = (const float*)d_in[5];

    const int DH   = in_sizes[3];            // 128
    const int DIN  = in_sizes[2] / DH;       // 64
    const int DOUT = in_sizes[5];            // 64
    const int N    = in_sizes[0] / DIN;      // 50000
    const int E    = in_sizes[1] / 2;        // 800000

    const long long* src = EI;
    const long long* dst = EI + E;

    float* ws   = (float*)d_ws;
    int    Na   = (N + 63) & ~63;
    float* dis  = ws;                        // N floats (deg -> rsqrt in place)
    float* bufA = ws + Na;                   // N*DH floats
    float* bufB = bufA + (size_t)N * DH;     // N*DH floats

    // --- symmetric normalization: dis = rsqrt(1 + indeg) ---
    fill1_kernel<<<cdiv(N, 256), 256, 0, stream>>>(dis, N);
    deg_edges_kernel<<<cdiv(E, 256), 256, 0, stream>>>(dst, dis, E);
    rsqrt_ip_kernel<<<cdiv(N, 256), 256, 0, stream>>>(dis, N);

    // --- layer 1: xw1 = X @ W1 (WMMA) -> bufA; agg1 init -> bufB; scatter ---
    gemm16_wmma_fused_kernel<<<cdiv(N, 16), 2 * DH, (size_t)16 * (DIN + 1) * sizeof(float), stream>>>(
        X, W1, /*preBias=*/nullptr, dis, bufA, bufB, N, DIN, DH);
    scatter_kernel<128><<<cdiv(E * (DH / 4), 256), 256, 0, stream>>>(bufA, src, dst, dis, bufB, E);

    // --- layer 2: xw2 = leaky(agg1+b1) @ W2 (WMMA, bias+relu fused in staging) ---
    float* xw2  = bufA;                      // N*DOUT floats (bufA dead after scatter1)
    float* agg2 = bufA + (size_t)N * DOUT;   // N*DOUT floats (second half of bufA)
    gemm16_wmma_fused_kernel<<<cdiv(N, 16), 2 * DOUT, (size_t)16 * (DH + 1) * sizeof(float), stream>>>(
        bufB, W2, /*preBias=*/b1, dis, xw2, agg2, N, DH, DOUT);
    scatter_kernel<64><<<cdiv(E * (DOUT / 4), 256), 256, 0, stream>>>(xw2, src, dst, dis, agg2, E);

    // --- final bias + leaky -> output ---
    bias_relu_kernel<64><<<cdiv(N * DOUT, 256), 256, 0, stream>>>(agg2, b2, (float*)d_out, N * DOUT);
}